// RbcNetwork_24970939859068
// MI455X (gfx1250) — compile-verified
//
#include <hip/hip_runtime.h>
#include <math.h>

#define N        80
#define NB       (N * N)     // 6400 batches
#define MAT      (N * N)     // 6400 elements per 80x80 matrix
#define LDB      82          // padded LDS row stride (floats) -> near conflict-free tile loads
#define PI_ITERS 25

typedef float v2f __attribute__((ext_vector_type(2)));
typedef float v8f __attribute__((ext_vector_type(8)));

__global__ __launch_bounds__(128) void rbc_zero_out(float* __restrict__ out) {
    if (threadIdx.x < N) out[threadIdx.x] = 0.0f;
}

__global__ __launch_bounds__(32)
void rbc_power_kernel(const float* __restrict__ r_zeros,
                      const float* __restrict__ r_const,
                      const float* __restrict__ traffic,
                      const float* __restrict__ wt,
                      const float* __restrict__ wr,
                      float* __restrict__ out)
{
    __shared__ float Blds[N * LDB];   // 80x80 matrix, padded rows (26240 B)
    __shared__ float vlds[N];         // current iterate v
    __shared__ float wlds[N];         // w = B v before normalization

    const int b    = blockIdx.x;      // batch index = a*80 + bb
    const int lane = threadIdx.x;     // wave32: one wave per batch, EXEC all-1s
    const int half = lane >> 4;       // 0: lanes 0-15, 1: lanes 16-31
    const int l15  = lane & 15;

    // ---- Stage 1: build B = sigmoid(weights_r)*r_zeros + r_const into LDS ----
    // Batch b owns contiguous 6400 floats of each 4D input. Coalesced float4 loads.
    const size_t  base = (size_t)b * MAT;
    const float4* z4 = (const float4*)(r_zeros + base);
    const float4* c4 = (const float4*)(r_const + base);
    const float4* w4 = (const float4*)(wr      + base);

    #pragma unroll 5
    for (int t = 0; t < 50; ++t) {
        const int e4 = t * 32 + lane;          // float4 index within batch
        const float4 x = w4[e4];
        const float4 z = z4[e4];
        const float4 c = c4[e4];
        float4 r;
        r.x = fmaf(1.0f / (1.0f + __expf(-x.x)), z.x, c.x);
        r.y = fmaf(1.0f / (1.0f + __expf(-x.y)), z.y, c.y);
        r.z = fmaf(1.0f / (1.0f + __expf(-x.z)), z.z, c.z);
        r.w = fmaf(1.0f / (1.0f + __expf(-x.w)), z.w, c.w);
        const int e   = e4 * 4;                // 80 % 4 == 0 -> float4 never crosses a row
        const int row = e / N;
        const int col = e - row * N;
        float* dst = &Blds[row * LDB + col];   // (row*82+col) even -> 8B-aligned b64 stores
        ((float2*)dst)[0] = make_float2(r.x, r.y);
        ((float2*)dst)[1] = make_float2(r.z, r.w);
    }

    // v0 = ones (matches reference exactly; 25 normalized iterations follow)
    vlds[lane]      = 1.0f;
    vlds[lane + 32] = 1.0f;
    if (lane < 16) vlds[lane + 64] = 1.0f;
    __syncthreads();

    const v8f vzero = {0.f, 0.f, 0.f, 0.f, 0.f, 0.f, 0.f, 0.f};

    // ---- Stage 2: 25 power iterations, all out of LDS via f32 WMMA ----
    for (int it = 0; it < PI_ITERS; ++it) {
        v8f acc[5];
        #pragma unroll
        for (int rc = 0; rc < 5; ++rc) acc[rc] = vzero;

        // D(16x16) = A(16x4 tile of B) x Bop(4x16 = v-chunk replicated over N)
        // K -> (VGPR, half) mapping mirrors the documented 16x4 A layout:
        //   VGPR0: K=0 (lanes 0-15), K=2 (lanes 16-31); VGPR1: K=1 / K=3.
        #pragma unroll
        for (int kc = 0; kc < 20; ++kc) {
            const float2 bv = *(const float2*)&vlds[4 * kc + 2 * half]; // LDS broadcast read
            v2f bop; bop[0] = bv.x; bop[1] = bv.y;
            #pragma unroll
            for (int rc = 0; rc < 5; ++rc) {
                const float2 at =
                    *(const float2*)&Blds[(16 * rc + l15) * LDB + 4 * kc + 2 * half];
                v2f aop; aop[0] = at.x; aop[1] = at.y;
                acc[rc] = __builtin_amdgcn_wmma_f32_16x16x4_f32(
                    false, aop, false, bop, (short)0, acc[rc], false, false);
            }
        }

        // D layout: VGPR j, lane 0 = w[16rc+j]; lane 16 = w[16rc+8+j]. Columns are
        // identical (v replicated), so lanes 0 and 16 spill the unique values.
        if (l15 == 0) {
            #pragma unroll
            for (int rc = 0; rc < 5; ++rc) {
                #pragma unroll
                for (int j = 0; j < 8; ++j)
                    wlds[16 * rc + 8 * half + j] = acc[rc][j];
            }
        }
        __syncthreads();

        // norm + normalize: w / (||w|| + 1e-12), exactly as the reference
        const float wa = wlds[lane];
        const float wb = wlds[lane + 32];
        const float wc = (lane < 16) ? wlds[lane + 64] : 0.0f;
        float p = wa * wa + wb * wb + wc * wc;
        #pragma unroll
        for (int off = 16; off > 0; off >>= 1) p += __shfl_xor(p, off, 32);
        const float inv = 1.0f / (sqrtf(p) + 1e-12f);

        vlds[lane]      = wa * inv;
        vlds[lane + 32] = wb * inv;
        if (lane < 16) vlds[lane + 64] = wc * inv;
        __syncthreads();
    }

    // ---- Stage 3: out[j] += T[a,bb] / v[a] * v[j]  (diag[a,bb] = V[a,bb,a] = v[a]) ----
    const int   a     = b / N;
    const float T     = wt[b] * traffic[b];
    const float scale = T / vlds[a];
    atomicAdd(&out[lane],      vlds[lane]      * scale);
    atomicAdd(&out[lane + 32], vlds[lane + 32] * scale);
    if (lane < 16) atomicAdd(&out[lane + 64], vlds[lane + 64] * scale);
}

extern "C" void kernel_launch(void* const* d_in, const int* in_sizes, int n_in,
                              void* d_out, int out_size, void* d_ws, size_t ws_size,
                              hipStream_t stream) {
    const float* r_zeros = (const float*)d_in[0];
    const float* r_const = (const float*)d_in[1];
    const float* traffic = (const float*)d_in[2];
    const float* wt      = (const float*)d_in[3];
    const float* wr      = (const float*)d_in[4];
    float* out = (float*)d_out;

    hipLaunchKernelGGL(rbc_zero_out, dim3(1), dim3(128), 0, stream, out);
    hipLaunchKernelGGL(rbc_power_kernel, dim3(NB), dim3(32), 0, stream,
                       r_zeros, r_const, traffic, wt, wr, out);
}